// TransformerConv2_2396591751945
// MI455X (gfx1250) — compile-verified
//
#include <hip/hip_runtime.h>
#include <math.h>

typedef __attribute__((ext_vector_type(2))) float v2f;
typedef __attribute__((ext_vector_type(8))) float v8f;

#define DH 16

// ---------------- helpers ----------------
__device__ __forceinline__ void atomicMaxF(float* addr, float val) {
  // signed-max for non-negative, unsigned-min for negative (IEEE ordering trick)
  if (val >= 0.0f) atomicMax((int*)addr, __float_as_int(val));
  else             atomicMin((unsigned int*)addr, __float_as_uint(val));
}

// ---------------- init kernels ----------------
__global__ void gt_init_layer(float* __restrict__ smax, float* __restrict__ denom,
                              float* __restrict__ agg, int N) {
  int tid = blockIdx.x * blockDim.x + threadIdx.x;
  if (tid < N) { smax[tid] = -INFINITY; denom[tid] = 0.0f; }
  if (tid < N * DH) agg[tid] = 0.0f;
}

__global__ void gt_init_pool(float* __restrict__ sums, float* __restrict__ cnt, int G) {
  int tid = blockIdx.x * blockDim.x + threadIdx.x;
  if (tid < G * DH) sums[tid] = 0.0f;
  if (tid < G) cnt[tid] = 0.0f;
}

// ---------------- fused q/k/v/skip projection via f32 WMMA ----------------
// One wave computes a 16-node x 16-feature tile for all four weight matrices.
// A: 16xK tile of X (f32).  B: Kx16 weights.  C init = bias broadcast.
// Requires N % 16 == 0 (holds: 100000 = 6250*16).
__global__ __launch_bounds__(256)
void gt_qkvs_wmma(const float* __restrict__ X, int Din,
                  const float* __restrict__ Wq, const float* __restrict__ bq,
                  const float* __restrict__ Wk, const float* __restrict__ bk,
                  const float* __restrict__ Wv, const float* __restrict__ bv,
                  const float* __restrict__ Ws, const float* __restrict__ bs,
                  float* __restrict__ Q, float* __restrict__ K,
                  float* __restrict__ V, float* __restrict__ S,
                  int nTiles) {
  int wave = blockIdx.x * (blockDim.x >> 5) + (threadIdx.x >> 5);
  if (wave >= nTiles) return;
  int lane = threadIdx.x & 31;
  int col  = lane & 15;   // A: M-row index; B/D: N-col index
  int hi   = lane >> 4;   // half-wave selector

  float bqv = bq[col], bkv = bk[col], bvv = bv[col], bsv = bs[col];
  v8f cq, ck, cv, cs;
#pragma unroll
  for (int j = 0; j < 8; ++j) { cq[j] = bqv; ck[j] = bkv; cv[j] = bvv; cs[j] = bsv; }

  int node0 = wave * 16;
  const float* arow = X + (size_t)(node0 + col) * (size_t)Din;

  for (int k0 = 0; k0 < Din; k0 += 4) {
    v2f a;                                   // A[row][k0+2*hi .. +1]
    a[0] = arow[k0 + 2 * hi];
    a[1] = arow[k0 + 2 * hi + 1];
    int r0 = (k0 + 2 * hi) * DH + col;       // B[k0+2*hi][col], next row +16
    v2f bqf, bkf, bvf, bsf;
    bqf[0] = Wq[r0]; bqf[1] = Wq[r0 + DH];
    bkf[0] = Wk[r0]; bkf[1] = Wk[r0 + DH];
    bvf[0] = Wv[r0]; bvf[1] = Wv[r0 + DH];
    bsf[0] = Ws[r0]; bsf[1] = Ws[r0 + DH];
    cq = __builtin_amdgcn_wmma_f32_16x16x4_f32(false, a, false, bqf, (short)0, cq, false, false);
    ck = __builtin_amdgcn_wmma_f32_16x16x4_f32(false, a, false, bkf, (short)0, ck, false, false);
    cv = __builtin_amdgcn_wmma_f32_16x16x4_f32(false, a, false, bvf, (short)0, cv, false, false);
    cs = __builtin_amdgcn_wmma_f32_16x16x4_f32(false, a, false, bsf, (short)0, cs, false, false);
  }

#pragma unroll
  for (int j = 0; j < 8; ++j) {              // D: VGPR j -> row j + 8*hi, col
    int idx = (node0 + j + 8 * hi) * DH + col;
    Q[idx] = cq[j]; K[idx] = ck[j]; V[idx] = cv[j]; S[idx] = cs[j];
  }
}

// ---------------- edge kernels (16 lanes per edge, lane = feature dim) ----------------
__global__ __launch_bounds__(256)
void gt_edge_score(const float* __restrict__ Q, const float* __restrict__ K,
                   const int* __restrict__ src, const int* __restrict__ dst,
                   float* __restrict__ score, float* __restrict__ smax, long long E) {
  long long tid = (long long)blockIdx.x * blockDim.x + threadIdx.x;
  long long e = tid >> 4;
  int c = (int)(tid & 15);
  if (e >= E) return;
  int s = src[e], d = dst[e];
  float p = Q[(size_t)d * DH + c] * K[(size_t)s * DH + c];
  p += __shfl_xor(p, 8);
  p += __shfl_xor(p, 4);
  p += __shfl_xor(p, 2);
  p += __shfl_xor(p, 1);
  if (c == 0) {
    float sc = p * 0.25f;                    // 1/sqrt(16)
    score[e] = sc;
    atomicMaxF(&smax[d], sc);
  }
}

__global__ __launch_bounds__(256)
void gt_edge_exp(float* __restrict__ score, const int* __restrict__ dst,
                 const float* __restrict__ smax, float* __restrict__ denom, long long E) {
  long long e = (long long)blockIdx.x * blockDim.x + threadIdx.x;
  if (e >= E) return;
  int d = dst[e];
  float ex = expf(score[e] - smax[d]);
  score[e] = ex;                             // reuse buffer: now holds exp
  atomicAdd(&denom[d], ex);
}

__global__ __launch_bounds__(256)
void gt_edge_agg(const float* __restrict__ score, const float* __restrict__ denom,
                 const float* __restrict__ V,
                 const int* __restrict__ src, const int* __restrict__ dst,
                 float* __restrict__ agg, long long E) {
  long long tid = (long long)blockIdx.x * blockDim.x + threadIdx.x;
  long long e = tid >> 4;
  int c = (int)(tid & 15);
  if (e >= E) return;
  int s = src[e], d = dst[e];
  float alpha = score[e] / (denom[d] + 1e-16f);
  atomicAdd(&agg[(size_t)d * DH + c], V[(size_t)s * DH + c] * alpha);
}

__global__ void gt_node_finish(const float* __restrict__ agg, const float* __restrict__ S,
                               float* __restrict__ H, int N) {
  int i = blockIdx.x * blockDim.x + threadIdx.x;
  if (i >= N * DH) return;
  float v = agg[i] + S[i];
  H[i] = v > 0.0f ? v : 0.0f;                // relu
}

// ---------------- global mean pool ----------------
__global__ void gt_pool(const float* __restrict__ H, const int* __restrict__ batch,
                        float* __restrict__ sums, float* __restrict__ cnt, int N) {
  int tid = blockIdx.x * blockDim.x + threadIdx.x;
  int n = tid >> 4, c = tid & 15;
  if (n >= N) return;
  int g = batch[n];
  atomicAdd(&sums[g * DH + c], H[(size_t)n * DH + c]);
  if (c == 0) atomicAdd(&cnt[g], 1.0f);
}

// ---------------- MLP head + log_softmax, one workgroup per graph ----------------
__device__ __forceinline__ float elu1(float z) { return z > 0.0f ? z : (expf(z) - 1.0f); }

__global__ __launch_bounds__(128)
void gt_mlp_head(const float* __restrict__ sums, const float* __restrict__ cnt,
                 const float* __restrict__ Wl1, const float* __restrict__ bl1,
                 const float* __restrict__ Wl2, const float* __restrict__ bl2,
                 const float* __restrict__ Wl3, const float* __restrict__ bl3,
                 float* __restrict__ out, int nClasses) {
  __shared__ float gv[DH];
  __shared__ float a1[64];
  __shared__ float a2[128];
  __shared__ float a3[16];
  int g = blockIdx.x, t = threadIdx.x;
  if (t < DH) gv[t] = sums[g * DH + t] / fmaxf(cnt[g], 1.0f);
  __syncthreads();
  if (t < 64) {
    float z = bl1[t];
#pragma unroll
    for (int i = 0; i < DH; ++i) z += gv[i] * Wl1[i * 64 + t];
    a1[t] = elu1(z);
  }
  __syncthreads();
  if (t < 128) {
    float z = bl2[t];
#pragma unroll 8
    for (int i = 0; i < 64; ++i) z += a1[i] * Wl2[i * 128 + t];
    a2[t] = elu1(z);
  }
  __syncthreads();
  if (t < nClasses) {
    float z = bl3[t];
#pragma unroll 8
    for (int i = 0; i < 128; ++i) z += a2[i] * Wl3[i * nClasses + t];
    a3[t] = elu1(z);
  }
  __syncthreads();
  if (t == 0) {
    float m = -INFINITY;
    for (int j = 0; j < nClasses; ++j) m = fmaxf(m, a3[j]);
    float ssum = 0.0f;
    for (int j = 0; j < nClasses; ++j) ssum += expf(a3[j] - m);
    float ls = logf(ssum);
    for (int j = 0; j < nClasses; ++j) out[g * nClasses + j] = a3[j] - m - ls;
  }
}

// ---------------- launch ----------------
extern "C" void kernel_launch(void* const* d_in, const int* in_sizes, int n_in,
                              void* d_out, int out_size, void* d_ws, size_t ws_size,
                              hipStream_t stream) {
  const float* x     = (const float*)d_in[0];
  const int*   ei    = (const int*)d_in[1];
  const int*   batch = (const int*)d_in[2];
  const float *Wq1 = (const float*)d_in[3],  *bq1 = (const float*)d_in[4];
  const float *Wk1 = (const float*)d_in[5],  *bk1 = (const float*)d_in[6];
  const float *Wv1 = (const float*)d_in[7],  *bv1 = (const float*)d_in[8];
  const float *Ws1 = (const float*)d_in[9],  *bs1 = (const float*)d_in[10];
  const float *Wq2 = (const float*)d_in[11], *bq2 = (const float*)d_in[12];
  const float *Wk2 = (const float*)d_in[13], *bk2 = (const float*)d_in[14];
  const float *Wv2 = (const float*)d_in[15], *bv2 = (const float*)d_in[16];
  const float *Ws2 = (const float*)d_in[17], *bs2 = (const float*)d_in[18];
  const float *Wl1 = (const float*)d_in[19], *bl1 = (const float*)d_in[20];
  const float *Wl2 = (const float*)d_in[21], *bl2 = (const float*)d_in[22];
  const float *Wl3 = (const float*)d_in[23], *bl3 = (const float*)d_in[24];
  float* out = (float*)d_out;

  const int N   = in_sizes[2];
  const long long E = (long long)in_sizes[1] / 2;
  const int Din = in_sizes[0] / N;
  const int nClasses = 10;
  const int G   = out_size / nClasses;
  const int* srcIdx = ei;
  const int* dstIdx = ei + E;

  // workspace carve-out (256B aligned slices)
  char* w = (char*)d_ws;
  auto carve = [&](size_t bytes) -> float* {
    float* p = (float*)w;
    w += (bytes + 255) & ~(size_t)255;
    return p;
  };
  const size_t nf = (size_t)N * DH * sizeof(float);
  float* Q     = carve(nf);
  float* Kb    = carve(nf);
  float* Vb    = carve(nf);
  float* Sb    = carve(nf);
  float* agg   = carve(nf);
  float* h1    = carve(nf);
  float* h2    = carve(nf);
  float* smax  = carve((size_t)N * sizeof(float));
  float* denom = carve((size_t)N * sizeof(float));
  float* score = carve((size_t)E * sizeof(float));
  float* sums  = carve((size_t)G * DH * sizeof(float));
  float* cnt   = carve((size_t)G * sizeof(float));

  const int TB = 256;
  const int nTiles   = (N + 15) / 16;                    // N==100000 -> exact 16-tiles
  const int qkvBlks  = (nTiles + 7) / 8;                 // 8 waves/block
  const int nodeBlks = (N * DH + TB - 1) / TB;
  const long long e16 = E * 16;
  const int edge16Blks = (int)((e16 + TB - 1) / TB);
  const int edgeBlks   = (int)((E + TB - 1) / TB);
  const int poolBlks   = (N * DH + TB - 1) / TB;
  const int poolInitBlks = (G * DH + TB - 1) / TB;

  // ---- layer 1 ----
  gt_init_layer<<<nodeBlks, TB, 0, stream>>>(smax, denom, agg, N);
  gt_qkvs_wmma<<<qkvBlks, TB, 0, stream>>>(x, Din, Wq1, bq1, Wk1, bk1, Wv1, bv1,
                                           Ws1, bs1, Q, Kb, Vb, Sb, nTiles);
  gt_edge_score<<<edge16Blks, TB, 0, stream>>>(Q, Kb, srcIdx, dstIdx, score, smax, E);
  gt_edge_exp<<<edgeBlks, TB, 0, stream>>>(score, dstIdx, smax, denom, E);
  gt_edge_agg<<<edge16Blks, TB, 0, stream>>>(score, denom, Vb, srcIdx, dstIdx, agg, E);
  gt_node_finish<<<nodeBlks, TB, 0, stream>>>(agg, Sb, h1, N);

  // ---- layer 2 ----
  gt_init_layer<<<nodeBlks, TB, 0, stream>>>(smax, denom, agg, N);
  gt_qkvs_wmma<<<qkvBlks, TB, 0, stream>>>(h1, DH, Wq2, bq2, Wk2, bk2, Wv2, bv2,
                                           Ws2, bs2, Q, Kb, Vb, Sb, nTiles);
  gt_edge_score<<<edge16Blks, TB, 0, stream>>>(Q, Kb, srcIdx, dstIdx, score, smax, E);
  gt_edge_exp<<<edgeBlks, TB, 0, stream>>>(score, dstIdx, smax, denom, E);
  gt_edge_agg<<<edge16Blks, TB, 0, stream>>>(score, denom, Vb, srcIdx, dstIdx, agg, E);
  gt_node_finish<<<nodeBlks, TB, 0, stream>>>(agg, Sb, h2, N);

  // ---- pool + head ----
  gt_init_pool<<<poolInitBlks, TB, 0, stream>>>(sums, cnt, G);
  gt_pool<<<poolBlks, TB, 0, stream>>>(h2, batch, sums, cnt, N);
  gt_mlp_head<<<G, 128, 0, stream>>>(sums, cnt, Wl1, bl1, Wl2, bl2, Wl3, bl3,
                                     out, nClasses);
}